// RPN_87703232184710
// MI455X (gfx1250) — compile-verified
//
#include <hip/hip_runtime.h>
#include <cstdint>
#include <cstddef>

#define AS1 __attribute__((address_space(1)))
#define AS3 __attribute__((address_space(3)))

#if defined(__has_builtin)
#  if __has_builtin(__builtin_amdgcn_global_load_async_to_lds_b32)
#    define HAVE_ASYNC_LDS_BUILTIN 1
#  endif
#  if __has_builtin(__builtin_amdgcn_s_wait_asynccnt)
#    define HAVE_WAIT_ASYNC_BUILTIN 1
#  endif
#endif

// gfx1250 async global->LDS copy of one DWORD per lane (ASYNCcnt tracked).
__device__ __forceinline__ void async_gld_to_lds_b32(const float* g, float* l) {
#ifdef HAVE_ASYNC_LDS_BUILTIN
    __builtin_amdgcn_global_load_async_to_lds_b32(
        (AS1 int*)(uintptr_t)g, (AS3 int*)l, 0, 0);
#else
    unsigned lds_off = (unsigned)(uintptr_t)(AS3 int*)l;   // LDS byte offset
    unsigned long long gaddr = (unsigned long long)(uintptr_t)g;
    asm volatile("global_load_async_to_lds_b32 %0, %1, off"
                 :: "v"(lds_off), "v"(gaddr) : "memory");
#endif
}

__device__ __forceinline__ void wait_asynccnt0() {
#ifdef HAVE_WAIT_ASYNC_BUILTIN
    __builtin_amdgcn_s_wait_asynccnt(0);
#else
    asm volatile("s_wait_asynccnt 0" ::: "memory");
#endif
}

namespace {
constexpr int   kN       = 64512;   // 128^2*3 + 64^2*3 + 32^2*3
constexpr int   kM       = 64;      // GT boxes per batch
constexpr int   kB       = 8;
constexpr int   kEndP3   = 49152;
constexpr int   kEndP4   = 61440;
constexpr float kSqrt2      = 1.41421356237309515f;
constexpr float kInvSqrt2   = 0.70710678118654752f;
constexpr float kScaleClamp = 3.3322045101752038f;  // log(224/8)
}

__global__ __launch_bounds__(256) void rpn_decode_match(
    const float* __restrict__ deltas,
    const float* __restrict__ gt_boxes,
    float* __restrict__ out_prop,
    float* __restrict__ out_match)
{
    __shared__ float s_gt[kM * 5];     // GT boxes for this batch (1.25 KB)
    __shared__ float s_area[kM];       // precomputed GT areas
    __shared__ float s_out[256 * 5];   // staging for coalesced matched stores (5 KB)

    const int tid = threadIdx.x;
    const int b   = blockIdx.y;
    const int n   = blockIdx.x * 256 + tid;   // grid.x*256 == kN exactly

    // ---- stage this batch's GT boxes into LDS with gfx1250 async loads ----
    const float* gtb = gt_boxes + (size_t)b * (kM * 5);
    async_gld_to_lds_b32(gtb + tid, &s_gt[tid]);                    // floats 0..255
    if (tid < kM * 5 - 256)
        async_gld_to_lds_b32(gtb + 256 + tid, &s_gt[256 + tid]);    // floats 256..319
    wait_asynccnt0();
    __syncthreads();

    // ---- precompute GT areas once per block ----
    if (tid < kM) {
        float g0 = s_gt[tid * 5 + 0], g1 = s_gt[tid * 5 + 1];
        float g2 = s_gt[tid * 5 + 2], g3 = s_gt[tid * 5 + 3];
        s_area[tid] = (g2 - g0) * (g3 - g1);
    }
    __syncthreads();

    // ---- anchor for global index n (analytic; bit-matches reference) ----
    int base, nside; float stride;
    if (n < kEndP3)      { base = 0;      nside = 128; stride = 8.f;  }
    else if (n < kEndP4) { base = kEndP3; nside = 64;  stride = 16.f; }
    else                 { base = kEndP4; nside = 32;  stride = 32.f; }
    int r   = n - base;
    int loc = r / 3;
    int ar  = r - loc * 3;          // aspect ratio index: 0->0.5, 1->1.0, 2->2.0
    int row = loc / nside;
    int col = loc - row * nside;
    float xc = (col + 0.5f) * stride;
    float yc = (row + 0.5f) * stride;
    float s4 = 4.f * stride;        // sqrt(area); power of two
    float wm = (ar == 0) ? kSqrt2 : ((ar == 1) ? 1.f : kInvSqrt2);
    float ws = s4 * wm;             // exact pow2 scaling of f32(sqrt2)
    float hs = (s4 * s4) / ws;      // = area / ws in f32, as reference

    float ax1 = xc - ws * 0.5f;
    float ay1 = yc - hs * 0.5f;
    float ax2 = xc + ws * 0.5f;
    float ay2 = yc + hs * 0.5f;

    // ---- decode deltas -> proposal box ----
    float4 dlt = reinterpret_cast<const float4*>(deltas)[(size_t)b * kN + n];
    float tw = fminf(dlt.z, kScaleClamp);
    float th = fminf(dlt.w, kScaleClamp);
    float pw = ax2 - ax1, ph = ay2 - ay1;
    float px = ax1 + pw * 0.5f;
    float py = ay2 - ph * 0.5f;     // reference uses anchors[:,3] - ph/2
    float bx = px + pw * dlt.x;
    float by = py + ph * dlt.y;
    float bw = pw * expf(tw);
    float bh = ph * expf(th);
    float x1 = bx - bw * 0.5f;
    float y1 = by - bh * 0.5f;
    float x2 = bx + bw * 0.5f;
    float y2 = by + bh * 0.5f;
    if (th < -1.0e7f) { x1 = y1 = x2 = y2 = -1.0e8f; }

    const size_t on = (size_t)b * kN + n;
    reinterpret_cast<float4*>(out_prop)[on] = make_float4(x1, y1, x2, y2);

    // ---- match against 64 GT boxes (LDS broadcast reads) ----
    float area1 = (x2 - x1) * (y2 - y1);
    float best  = -2.f;
    int   bidx  = 0;
    #pragma unroll 8
    for (int g = 0; g < kM; ++g) {
        float g0 = s_gt[g * 5 + 0], g1 = s_gt[g * 5 + 1];
        float g2 = s_gt[g * 5 + 2], g3 = s_gt[g * 5 + 3];
        float g4 = s_gt[g * 5 + 4];
        float m = -1.f;
        if (g4 != -1.f) {           // valid GT
            float w = fminf(x2, g2) - fmaxf(x1, g0);
            float h = fminf(y2, g3) - fmaxf(y1, g1);
            w = fmaxf(w, 0.f);
            h = fmaxf(h, 0.f);
            float inter = w * h;
            m = inter / (area1 + s_area[g] - inter);
        }
        if (m > best) { best = m; bidx = g; }   // strict '>' = first-max (jnp argmax)
    }

    float o0, o1, o2, o3, o4;
    if (best <= 0.3f) {
        o0 = o1 = o2 = o3 = o4 = -1.f;
    } else if (best < 0.6f) {
        o0 = o1 = o2 = o3 = o4 = -1.0e8f;
    } else {
        o0 = s_gt[bidx * 5 + 0]; o1 = s_gt[bidx * 5 + 1];
        o2 = s_gt[bidx * 5 + 2]; o3 = s_gt[bidx * 5 + 3];
        o4 = s_gt[bidx * 5 + 4];
    }

    // ---- stage matched rows in LDS, then fully-coalesced b128 stores ----
    s_out[tid * 5 + 0] = o0;
    s_out[tid * 5 + 1] = o1;
    s_out[tid * 5 + 2] = o2;
    s_out[tid * 5 + 3] = o3;
    s_out[tid * 5 + 4] = o4;
    __syncthreads();

    // Block's matched region: 256 rows * 5 floats = 320 float4 (16B aligned:
    // byte offset is a multiple of 5120).
    float4* dst = reinterpret_cast<float4*>(
        out_match + ((size_t)b * kN + (size_t)blockIdx.x * 256) * 5);
    const float4* src = reinterpret_cast<const float4*>(s_out);
    dst[tid] = src[tid];                         // float4 0..255
    if (tid < 64) dst[256 + tid] = src[256 + tid];  // float4 256..319
}

extern "C" void kernel_launch(void* const* d_in, const int* in_sizes, int n_in,
                              void* d_out, int out_size, void* d_ws, size_t ws_size,
                              hipStream_t stream)
{
    // setup_inputs order: locations_p3, locations_p4, locations_p5, deltas, gt_boxes
    const float* deltas   = (const float*)d_in[3];   // (8, 64512, 4)
    const float* gt_boxes = (const float*)d_in[4];   // (8, 64, 5)
    float* out_prop  = (float*)d_out;                        // (8, 64512, 4)
    float* out_match = out_prop + (size_t)kB * kN * 4;       // (8, 64512, 5)

    dim3 grid(kN / 256, kB, 1);      // 252 x 8 blocks, 8 wave32 per block
    rpn_decode_match<<<grid, dim3(256, 1, 1), 0, stream>>>(
        deltas, gt_boxes, out_prop, out_match);
}